// _EuclideanCodebook_32323923870089
// MI455X (gfx1250) — compile-verified
//
#include <hip/hip_runtime.h>

#define NROWS   65536
#define DIM     128
#define KCODES  1024
#define DECAY_F 0.99f
#define OMD_F   0.01f
#define EPS_F   1e-5f

typedef __attribute__((ext_vector_type(2))) float v2f;
typedef __attribute__((ext_vector_type(8))) float v8f;

// ---------------- zero small workspace head (total + counts) ----------------
__global__ __launch_bounds__(256) void k_zero(float* __restrict__ p, int n) {
    int i = blockIdx.x * 256 + threadIdx.x;
    if (i < n) p[i] = 0.0f;
}

// ---------------- ||e_k||^2 : one wave per code ----------------
__global__ __launch_bounds__(256) void k_enorm(const float* __restrict__ embed,
                                               float* __restrict__ enorm) {
    int wave = (blockIdx.x * 256 + threadIdx.x) >> 5;   // 0..1023
    int lane = threadIdx.x & 31;
    const float4* row = (const float4*)(embed + (size_t)wave * DIM);
    float4 v = row[lane];                               // 128 floats / 32 lanes
    float s = v.x * v.x + v.y * v.y + v.z * v.z + v.w * v.w;
    #pragma unroll
    for (int m = 16; m > 0; m >>= 1) s += __shfl_xor(s, m, 32);
    if (lane == 0) enorm[wave] = s;
}

// ---------------- fused WMMA GEMM + row argmax + code histogram ----------------
// grid 512 x 256: each block = 8 waves, each wave owns 16 rows and sweeps all
// 64 code tiles with V_WMMA_F32_16X16X4_F32, argmax kept in C fragments.
__global__ __launch_bounds__(256) void k_argmax(const float* __restrict__ x,
                                                const float* __restrict__ embed,
                                                const float* __restrict__ enorm,
                                                int* __restrict__ ind_i,
                                                float* __restrict__ ind_f,
                                                int* __restrict__ cnt) {
    const int lane = threadIdx.x & 31;
    const int wv   = threadIdx.x >> 5;
    const int r16  = lane & 15;            // lane within 16-lane half
    const int half = lane >> 4;            // 0 or 1
    const int kh   = half << 1;            // K offset 0 or 2 in A/B fragments
    const int rowBase = blockIdx.x * 128 + wv * 16;

    // Preload A fragments (16 rows x 128 dims, reused across all 64 tiles).
    v2f a[32];
    const float* xrow = x + (size_t)(rowBase + r16) * DIM + kh;
    #pragma unroll
    for (int kk = 0; kk < 32; ++kk)
        a[kk] = *(const v2f*)(xrow + 4 * kk);

    float best[8];
    int   bidx[8];
    #pragma unroll
    for (int r = 0; r < 8; ++r) { best[r] = -3.4e38f; bidx[r] = 0; }

    for (int t = 0; t < 64; ++t) {
        const int codeBase = t * 16;
        const float* erow = embed + (size_t)(codeBase + r16) * DIM + kh;
        // prefetch next tile's B stream (global_prefetch_b8)
        const int nt = (t + 1) & 63;
        __builtin_prefetch(embed + (size_t)(nt * 16 + r16) * DIM + kh, 0, 3);

        v8f c = {};
        #pragma unroll
        for (int kc = 0; kc < 32; kc += 8) {     // chunk B to bound registers
            v2f b[8];
            #pragma unroll
            for (int j = 0; j < 8; ++j)
                b[j] = *(const v2f*)(erow + 4 * (kc + j));
            #pragma unroll
            for (int j = 0; j < 8; ++j)
                c = __builtin_amdgcn_wmma_f32_16x16x4_f32(
                        false, a[kc + j], false, b[j], (short)0, c, false, false);
        }
        // score = 2*dot - ||e||^2 ; C layout: lane = col, VGPR r = row r / r+8
        float en   = enorm[codeBase + r16];
        int   cand = codeBase + r16;
        #pragma unroll
        for (int r = 0; r < 8; ++r) {
            float s = 2.0f * c[r] - en;
            if (s > best[r]) { best[r] = s; bidx[r] = cand; }   // strict > => first-index ties
        }
    }

    // Reduce across each 16-lane half (masks < 16 stay inside the half).
    #pragma unroll
    for (int r = 0; r < 8; ++r) {
        float v = best[r]; int i = bidx[r];
        #pragma unroll
        for (int m = 8; m >= 1; m >>= 1) {
            float ov = __shfl_xor(v, m, 32);
            int   oi = __shfl_xor(i, m, 32);
            if (ov > v || (ov == v && oi < i)) { v = ov; i = oi; }
        }
        if (r16 == 0) {
            int row = rowBase + r + (half << 3);   // half 0 -> rows r, half 1 -> rows r+8
            ind_i[row] = i;
            ind_f[row] = (float)i;
            atomicAdd(&cnt[i], 1);                 // code histogram (65K atomics total)
        }
    }
}

// ---------------- single-wave exclusive scan over 1024 counts ----------------
__global__ __launch_bounds__(32) void k_scan(const int* __restrict__ cnt,
                                             int* __restrict__ offsets,
                                             int* __restrict__ cursor) {
    int lane = threadIdx.x;                // 0..31, one wave
    int base = lane * 32;
    int s = 0;
    #pragma unroll
    for (int j = 0; j < 32; ++j) s += cnt[base + j];
    int incl = s;
    #pragma unroll
    for (int m = 1; m < 32; m <<= 1) {
        int v = __shfl_up(incl, m, 32);
        if (lane >= m) incl += v;
    }
    int run = incl - s;                    // exclusive prefix of this lane's chunk
    for (int j = 0; j < 32; ++j) {
        offsets[base + j] = run;
        cursor[base + j]  = run;
        run += cnt[base + j];
    }
}

// ---------------- scatter row ids into per-code lists ----------------
__global__ __launch_bounds__(256) void k_scatter_rows(const int* __restrict__ ind,
                                                      int* __restrict__ cursor,
                                                      int* __restrict__ rowlist) {
    int i = blockIdx.x * 256 + threadIdx.x;     // 0..65535
    int k = ind[i];
    int pos = atomicAdd(&cursor[k], 1);
    rowlist[pos] = i;
}

// ---------------- cluster-size EMA + total ----------------
__global__ __launch_bounds__(256) void k_cluster(const float* __restrict__ cs,
                                                 const int* __restrict__ cnt,
                                                 float* __restrict__ new_cs,
                                                 float* __restrict__ total) {
    int i = blockIdx.x * 256 + threadIdx.x;     // 0..1023
    float v = cs[i] * DECAY_F + OMD_F * (float)cnt[i];
    new_cs[i] = v;
    float s = v;
    #pragma unroll
    for (int m = 16; m > 0; m >>= 1) s += __shfl_xor(s, m, 32);
    if ((threadIdx.x & 31) == 0) atomicAdd(total, s);
}

// ---------------- per-code: segment sum of x + EMA + normalize + quantize ----
// grid 1024 x 128: block b owns code k=b; thread d owns dim d.
__global__ __launch_bounds__(128) void k_code_sum(const float* __restrict__ x,
                                                  const float* __restrict__ embed,
                                                  const float* __restrict__ eavg,
                                                  const int* __restrict__ offsets,
                                                  const int* __restrict__ cnt,
                                                  const int* __restrict__ rowlist,
                                                  const float* __restrict__ new_cs,
                                                  const float* __restrict__ total,
                                                  float* __restrict__ new_eavg,
                                                  float* __restrict__ new_embed,
                                                  float* __restrict__ quant) {
    const int k   = blockIdx.x;
    const int d   = threadIdx.x;
    const int beg = offsets[k];
    const int n   = cnt[k];

    float ev  = embed[(size_t)k * DIM + d];     // this code's vector (register-resident)
    float acc = 0.0f;
    int j = 0;
    for (; j + 4 <= n; j += 4) {                // batch loads to hide latency
        int r0 = rowlist[beg + j + 0];
        int r1 = rowlist[beg + j + 1];
        int r2 = rowlist[beg + j + 2];
        int r3 = rowlist[beg + j + 3];
        float x0 = x[(size_t)r0 * DIM + d];
        float x1 = x[(size_t)r1 * DIM + d];
        float x2 = x[(size_t)r2 * DIM + d];
        float x3 = x[(size_t)r3 * DIM + d];
        quant[(size_t)r0 * DIM + d] = ev;
        quant[(size_t)r1 * DIM + d] = ev;
        quant[(size_t)r2 * DIM + d] = ev;
        quant[(size_t)r3 * DIM + d] = ev;
        acc += x0; acc += x1; acc += x2; acc += x3;
    }
    for (; j < n; ++j) {
        int r = rowlist[beg + j];
        acc += x[(size_t)r * DIM + d];
        quant[(size_t)r * DIM + d] = ev;
    }

    float na = eavg[(size_t)k * DIM + d] * DECAY_F + OMD_F * acc;
    new_eavg[(size_t)k * DIM + d] = na;
    float tot = *total;
    float sm  = (new_cs[k] + EPS_F) / (tot + (float)KCODES * EPS_F) * tot;
    new_embed[(size_t)k * DIM + d] = na / sm;
}

extern "C" void kernel_launch(void* const* d_in, const int* in_sizes, int n_in,
                              void* d_out, int out_size, void* d_ws, size_t ws_size,
                              hipStream_t stream) {
    (void)in_sizes; (void)n_in; (void)out_size; (void)ws_size;
    const float* x     = (const float*)d_in[0];   // [8,8192,128]
    const float* embed = (const float*)d_in[1];   // [1024,128]
    const float* cs    = (const float*)d_in[2];   // [1024]
    const float* eavg  = (const float*)d_in[3];   // [1024,128]

    float* out     = (float*)d_out;
    float* quant   = out;                         // 8388608
    float* ind_f   = out + 8388608;               // 65536
    float* new_cs  = out + 8454144;               // 1024
    float* new_avg = out + 8455168;               // 131072
    float* new_emb = out + 8586240;               // 131072

    float* ws      = (float*)d_ws;                // dword layout:
    float* total   = ws;                          // [0]        total
    int*   cnt     = (int*)(ws + 1);              // [1,1025)   per-code counts
    int*   offsets = (int*)(ws + 1025);           // [1025,2049)
    int*   cursor  = (int*)(ws + 2049);           // [2049,3073)
    int*   rowlist = (int*)(ws + 3073);           // [3073,68609)
    int*   ind_i   = (int*)(ws + 68609);          // [68609,134145)
    float* enorm   = ws + 134145;                 // [134145,135169)

    k_zero         <<<(1025 + 255) / 256, 256, 0, stream>>>(ws, 1025);
    k_enorm        <<<128, 256, 0, stream>>>(embed, enorm);
    k_argmax       <<<512, 256, 0, stream>>>(x, embed, enorm, ind_i, ind_f, cnt);
    k_scan         <<<1, 32, 0, stream>>>(cnt, offsets, cursor);
    k_scatter_rows <<<256, 256, 0, stream>>>(ind_i, cursor, rowlist);
    k_cluster      <<<4, 256, 0, stream>>>(cs, cnt, new_cs, total);
    k_code_sum     <<<1024, 128, 0, stream>>>(x, embed, eavg, offsets, cnt, rowlist,
                                              new_cs, total, new_avg, new_emb, quant);
}